// Net_80255758893589
// MI455X (gfx1250) — compile-verified
//
#include <hip/hip_runtime.h>
#include <hip/hip_bf16.h>

#define N_NODES  100000
#define N_EDGES  3200000
#define N_GRAPHS 64
#define F        16

typedef float v2f __attribute__((ext_vector_type(2)));
typedef float v8f __attribute__((ext_vector_type(8)));

// ---------------- utility ----------------
__global__ void fill_zero_f(float* __restrict__ p, int n) {
    int i = blockIdx.x * blockDim.x + threadIdx.x;
    if (i < n) p[i] = 0.0f;
}

// deg over row (source), cnt over col (target)
__global__ void edge_degrees(const int* __restrict__ row, const int* __restrict__ col,
                             float* __restrict__ deg, float* __restrict__ cnt, int E) {
    int e = blockIdx.x * blockDim.x + threadIdx.x;
    if (e < E) {
        atomicAdd(&deg[row[e]], 1.0f);
        atomicAdd(&cnt[col[e]], 1.0f);
    }
}

__global__ void batch_counts(const int* __restrict__ batch, float* __restrict__ gcnt, int n) {
    int i = blockIdx.x * blockDim.x + threadIdx.x;
    if (i < n) atomicAdd(&gcnt[batch[i]], 1.0f);
}

__global__ void make_dinv(const float* __restrict__ deg, float* __restrict__ dinv, int n) {
    int i = blockIdx.x * blockDim.x + threadIdx.x;
    if (i < n) {
        float d = deg[i];
        dinv[i] = (d > 0.0f) ? rsqrtf(d) : 0.0f;
    }
}

// ---------------- layer-1 linear: x[N,3] @ W[3,16] via V_WMMA_F32_16X16X4_F32 ----------------
// A is 16x4 (K padded from 3 to 4 with zeros), B is 4x16, C/D 16x16 f32.
__global__ void lin3_wmma(const float* __restrict__ x,
                          const float* __restrict__ W1, const float* __restrict__ bias1,
                          const float* __restrict__ W2,
                          float* __restrict__ center, float* __restrict__ xj, int ntiles) {
    int wave = (blockIdx.x * blockDim.x + threadIdx.x) >> 5;   // one 16-row tile per wave
    int lane = threadIdx.x & 31;
    if (wave >= ntiles) return;                                // wave-uniform: EXEC stays full

    int m    = lane & 15;          // row within tile (A) / col N (B, C/D)
    int half = lane >> 4;          // 0: K 0..1, 1: K 2..3
    int kb   = half * 2;
    int node = wave * 16 + m;

    // A tile 16x4 (K=3 padded): lane holds X[m][kb], X[m][kb+1]
    v2f a;
    a.x = (kb + 0 < 3) ? x[node * 3 + kb + 0] : 0.0f;
    a.y = (kb + 1 < 3) ? x[node * 3 + kb + 1] : 0.0f;

    // B tiles 4x16: lane n holds W[kb][n], W[kb+1][n]  (row 3 padded with zeros)
    int n = m;
    v2f bw1, bw2;
    bw1.x = (kb + 0 < 3) ? W1[(kb + 0) * F + n] : 0.0f;
    bw1.y = (kb + 1 < 3) ? W1[(kb + 1) * F + n] : 0.0f;
    bw2.x = (kb + 0 < 3) ? W2[(kb + 0) * F + n] : 0.0f;
    bw2.y = (kb + 1 < 3) ? W2[(kb + 1) * F + n] : 0.0f;

    v8f c0 = {};
    v8f c1 = {};
    c0 = __builtin_amdgcn_wmma_f32_16x16x4_f32(false, a, false, bw1, (short)0, c0, false, false);
    c1 = __builtin_amdgcn_wmma_f32_16x16x4_f32(false, a, false, bw2, (short)0, c1, false, false);

    float bv = bias1[n];
#pragma unroll
    for (int v = 0; v < 8; ++v) {
        int r = wave * 16 + v + half * 8;   // C/D layout: VGPR v -> row v + 8*(lane/16)
        center[r * F + n] = c0[v] + bv;
        xj[r * F + n]     = c1[v];
    }
}

// ---------------- layer-2 linear: h[N,16] @ W[16,16], 4 K-chunks of 4 ----------------
__global__ void lin16_wmma(const float* __restrict__ h,
                           const float* __restrict__ W1, const float* __restrict__ bias1,
                           const float* __restrict__ W2,
                           float* __restrict__ center, float* __restrict__ xj, int ntiles) {
    int wave = (blockIdx.x * blockDim.x + threadIdx.x) >> 5;
    int lane = threadIdx.x & 31;
    if (wave >= ntiles) return;

    int m    = lane & 15;
    int half = lane >> 4;
    int kb   = half * 2;
    int n    = m;
    const float* hrow = h + (wave * 16 + m) * F;

    v8f c0 = {};
    v8f c1 = {};
#pragma unroll
    for (int kc = 0; kc < 4; ++kc) {
        int k0 = kc * 4 + kb;
        v2f a;   a.x   = hrow[k0];            a.y   = hrow[k0 + 1];
        v2f b1v; b1v.x = W1[k0 * F + n];      b1v.y = W1[(k0 + 1) * F + n];
        v2f b2v; b2v.x = W2[k0 * F + n];      b2v.y = W2[(k0 + 1) * F + n];
        c0 = __builtin_amdgcn_wmma_f32_16x16x4_f32(false, a, false, b1v, (short)0, c0, false, false);
        c1 = __builtin_amdgcn_wmma_f32_16x16x4_f32(false, a, false, b2v, (short)0, c1, false, false);
    }

    float bv = bias1[n];
#pragma unroll
    for (int v = 0; v < 8; ++v) {
        int r = wave * 16 + v + half * 8;
        center[r * F + n] = c0[v] + bv;
        xj[r * F + n]     = c1[v];
    }
}

// ---------------- edge scatter: one thread per (edge, feature) ----------------
__global__ void edge_scatter(const int* __restrict__ row, const int* __restrict__ col,
                             const float* __restrict__ dinv, const float* __restrict__ xj,
                             float* __restrict__ aggr, int E) {
    unsigned tid = blockIdx.x * blockDim.x + threadIdx.x;
    int e = tid >> 4;
    int k = tid & 15;
    if (e >= E) return;
    int r = row[e];
    int c = col[e];
    float norm = dinv[r] * dinv[c];
    float v = norm * xj[r * F + k];
    atomicAdd(&aggr[c * F + k], v);
}

// ---------------- h = relu(center + aggr/max(cnt,1)) ----------------
__global__ void finish_layer(const float* __restrict__ center, const float* __restrict__ aggr,
                             const float* __restrict__ cnt, float* __restrict__ hout, int n) {
    unsigned tid = blockIdx.x * blockDim.x + threadIdx.x;
    int i = tid >> 4;   // node
    int k = tid & 15;
    if (i >= n) return;
    float s = aggr[i * F + k] / fmaxf(cnt[i], 1.0f);
    hout[i * F + k] = fmaxf(center[i * F + k] + s, 0.0f);
}

// ---------------- finish layer 2 + fused graph-mean pooling accumulation ----------------
__global__ void finish_layer_pool(const float* __restrict__ center, const float* __restrict__ aggr,
                                  const float* __restrict__ cnt, const int* __restrict__ batch,
                                  float* __restrict__ gsum, int n) {
    unsigned tid = blockIdx.x * blockDim.x + threadIdx.x;
    int i = tid >> 4;
    int k = tid & 15;
    if (i >= n) return;
    float s = aggr[i * F + k] / fmaxf(cnt[i], 1.0f);
    float h = fmaxf(center[i * F + k] + s, 0.0f);
    atomicAdd(&gsum[batch[i] * F + k], h);
}

// ---------------- head: g = gsum/cnt; relu(g@l1W+b); @l2W+b -> out[64,2] ----------------
__global__ void head_kernel(const float* __restrict__ gsum, const float* __restrict__ gcnt,
                            const float* __restrict__ l1W, const float* __restrict__ l1b,
                            const float* __restrict__ l2W, const float* __restrict__ l2b,
                            float* __restrict__ out) {
    int g = threadIdx.x;
    if (g >= N_GRAPHS) return;
    float inv = 1.0f / fmaxf(gcnt[g], 1.0f);
    float gv[F];
#pragma unroll
    for (int k = 0; k < F; ++k) gv[k] = gsum[g * F + k] * inv;
    float a[F];
#pragma unroll
    for (int j = 0; j < F; ++j) {
        float s = l1b[j];
#pragma unroll
        for (int k = 0; k < F; ++k) s += gv[k] * l1W[k * F + j];
        a[j] = fmaxf(s, 0.0f);
    }
#pragma unroll
    for (int j = 0; j < 2; ++j) {
        float s = l2b[j];
#pragma unroll
        for (int k = 0; k < F; ++k) s += a[k] * l2W[k * 2 + j];
        out[g * 2 + j] = s;
    }
}

extern "C" void kernel_launch(void* const* d_in, const int* in_sizes, int n_in,
                              void* d_out, int out_size, void* d_ws, size_t ws_size,
                              hipStream_t stream) {
    const float* x     = (const float*)d_in[0];
    const int*   ei    = (const int*)d_in[1];
    const int*   batch = (const int*)d_in[2];
    const float* c1W1  = (const float*)d_in[3];
    const float* c1b1  = (const float*)d_in[4];
    const float* c1W2  = (const float*)d_in[5];
    const float* c2W1  = (const float*)d_in[6];
    const float* c2b1  = (const float*)d_in[7];
    const float* c2W2  = (const float*)d_in[8];
    const float* l1W   = (const float*)d_in[9];
    const float* l1b   = (const float*)d_in[10];
    const float* l2W   = (const float*)d_in[11];
    const float* l2b   = (const float*)d_in[12];
    float* out = (float*)d_out;

    const int* row = ei;             // edge_index[0]
    const int* col = ei + N_EDGES;   // edge_index[1]

    float* ws   = (float*)d_ws;
    float* f0   = ws;                        // N*16
    float* f1   = ws + (size_t)N_NODES * F;  // N*16
    float* f2   = ws + (size_t)2 * N_NODES * F;
    float* deg  = ws + (size_t)3 * N_NODES * F;  // N
    float* cnt  = deg + N_NODES;                 // N
    float* dinv = cnt + N_NODES;                 // N
    float* gsum = dinv + N_NODES;                // 64*16
    float* gcnt = gsum + N_GRAPHS * F;           // 64

    const int B = 256;
    const int nNF    = N_NODES * F;                 // 1,600,000
    const int ntiles = N_NODES / 16;                // 6250 (exact)
    const int gTiles = (ntiles * 32 + B - 1) / B;   // wave-per-tile grids
    const int gNodes = (N_NODES + B - 1) / B;
    const int gEdges = (N_EDGES + B - 1) / B;
    const int gNF    = nNF / B;                     // 6250 exact
    const int gEF    = (N_EDGES * 16) / B;          // 200000 exact

    // zero deg, cnt, dinv, gsum, gcnt (contiguous)
    int statN = 3 * N_NODES + N_GRAPHS * F + N_GRAPHS;
    fill_zero_f<<<(statN + B - 1) / B, B, 0, stream>>>(deg, statN);

    edge_degrees<<<gEdges, B, 0, stream>>>(row, col, deg, cnt, N_EDGES);
    batch_counts<<<gNodes, B, 0, stream>>>(batch, gcnt, N_NODES);
    make_dinv<<<gNodes, B, 0, stream>>>(deg, dinv, N_NODES);

    // ---- layer 1 ----
    lin3_wmma<<<gTiles, B, 0, stream>>>(x, c1W1, c1b1, c1W2, f0 /*center*/, f1 /*xj*/, ntiles);
    fill_zero_f<<<gNF, B, 0, stream>>>(f2, nNF);
    edge_scatter<<<gEF, B, 0, stream>>>(row, col, dinv, f1, f2, N_EDGES);
    finish_layer<<<gNF, B, 0, stream>>>(f0, f2, cnt, f0 /*h1 in place*/, N_NODES);

    // ---- layer 2 ----
    lin16_wmma<<<gTiles, B, 0, stream>>>(f0, c2W1, c2b1, c2W2, f1 /*center*/, f2 /*xj*/, ntiles);
    fill_zero_f<<<gNF, B, 0, stream>>>(f0, nNF);
    edge_scatter<<<gEF, B, 0, stream>>>(row, col, dinv, f2, f0, N_EDGES);
    finish_layer_pool<<<gNF, B, 0, stream>>>(f1, f0, cnt, batch, gsum, N_NODES);

    // ---- head ----
    head_kernel<<<1, 64, 0, stream>>>(gsum, gcnt, l1W, l1b, l2W, l2b, out);
}